// BinaryAdjustDiceLoss_29446295781636
// MI455X (gfx1250) — compile-verified
//
#include <hip/hip_runtime.h>
#include <stdint.h>
#include <stddef.h>

#define B_SAMPLES 16
#define N_PER_SAMPLE (1u << 20)
#define TPB 256
#define VEC 4
#define TILE (TPB * VEC)                                   // 1024 elements
#define TILES_PER_BLOCK 8
#define BLOCKS_PER_SAMPLE (N_PER_SAMPLE / (TILE * TILES_PER_BLOCK))  // 128
#define SMOOTH_F 1e-4f

__device__ __constant__ float c_ratios[8] = {
    0.317f, 0.329f, 0.326f, 0.115f, 0.701f, 0.367f, 1.22f, 0.241f};

// Order-preserving float -> u32 key (ascending float == ascending key).
__device__ __forceinline__ uint32_t f2key(float f) {
  uint32_t u = __float_as_uint(f);
  return (u & 0x80000000u) ? ~u : (u | 0x80000000u);
}

// ---- CDNA5 async global -> LDS data mover (ASYNCcnt path) ----
__device__ __forceinline__ void async_load_b128(uint32_t lds_off, const float* g) {
  asm volatile("global_load_async_to_lds_b128 %0, %1, off"
               :: "v"(lds_off), "v"((unsigned long long)(uintptr_t)g)
               : "memory");
}
__device__ __forceinline__ void wait_async_le2() {
  asm volatile("s_wait_asynccnt 2" ::: "memory");
}
__device__ __forceinline__ void wait_async_le1() {
  asm volatile("s_wait_asynccnt 1" ::: "memory");
}
__device__ __forceinline__ void wait_async_le0() {
  asm volatile("s_wait_asynccnt 0" ::: "memory");
}

// ---------------------------------------------------------------------------
// Histogram passes.
// LEVEL 0: bits[31:20] of neg keys + pos count (+ packed neg-mask if MOPT).
// LEVEL 1: bits[19:8] restricted to selected coarse bin.
// LEVEL 2: bits[7:0] restricted to selected 24-bit prefix.
// MOPT (LEVEL>0): read 1-byte-per-4-elements neg mask instead of target.
// ---------------------------------------------------------------------------
template <int LEVEL, int MOPT>
__global__ __launch_bounds__(TPB)
void hist_pass(const float* __restrict__ inp, const float* __restrict__ tgt,
               uint8_t* __restrict__ negMask,
               uint32_t* __restrict__ ghist, uint32_t* __restrict__ posCount,
               const uint32_t* __restrict__ prefix,
               const uint32_t* __restrict__ forceAll,
               const uint32_t* __restrict__ doneF) {
  const uint32_t BINS = (LEVEL == 2) ? 256u : 4096u;
  __shared__ __align__(16) float ibuf[2][TILE];
  __shared__ __align__(16) float tbuf[2][TILE];   // unused in masked refine
  __shared__ uint32_t lhist[4096];
  __shared__ uint32_t lpos;

  const int b = blockIdx.y;
  if (LEVEL > 0) {
    if (doneF[b] | forceAll[b]) return;   // threshold already resolved
  }
  const uint32_t pfx = (LEVEL > 0) ? prefix[b] : 0u;

  for (uint32_t i = threadIdx.x; i < BINS; i += TPB) lhist[i] = 0u;
  if (threadIdx.x == 0) lpos = 0u;
  __syncthreads();

  const int tid = threadIdx.x;
  const size_t base = (size_t)b * N_PER_SAMPLE +
                      (size_t)blockIdx.x * (size_t)(TILE * TILES_PER_BLOCK);

  auto issue = [&](int stage, int tile) {
    const float* gi = inp + base + (size_t)tile * TILE + tid * VEC;
    async_load_b128((uint32_t)(uintptr_t)&ibuf[stage][tid * VEC], gi);
    if (LEVEL == 0 || !MOPT) {
      const float* gt = tgt + base + (size_t)tile * TILE + tid * VEC;
      async_load_b128((uint32_t)(uintptr_t)&tbuf[stage][tid * VEC], gt);
    }
  };

  uint32_t posc = 0;
  issue(0, 0);
  for (int tile = 0; tile < TILES_PER_BLOCK; ++tile) {
    const int s = tile & 1;
    uint32_t mb = 0xFu;
    if (LEVEL > 0 && MOPT)   // mask byte for this thread's 4 elements
      mb = negMask[base / 4 + (size_t)tile * (TILE / 4) + tid];
    if (tile + 1 < TILES_PER_BLOCK) {
      issue(s ^ 1, tile + 1);
      if (LEVEL == 0 || !MOPT) wait_async_le2(); else wait_async_le1();
    } else {
      wait_async_le0();
    }
    volatile const float* iv = &ibuf[s][0];
    volatile const float* tv = &tbuf[s][0];
    uint32_t nib = 0;
#pragma unroll
    for (int e = 0; e < VEC; ++e) {
      const float x = iv[tid * VEC + e];
      if (LEVEL == 0) {
        const float t = tv[tid * VEC + e];
        if (t > 0.5f) {
          ++posc;
        } else {
          nib |= 1u << e;
          atomicAdd(&lhist[f2key(x) >> 20], 1u);
        }
      } else {
        bool neg;
        if (MOPT) {
          neg = (mb >> e) & 1u;
        } else {
          const float t = tv[tid * VEC + e];
          neg = !(t > 0.5f);
        }
        if (neg) {
          const uint32_t key = f2key(x);
          if (LEVEL == 1) {
            if ((key >> 20) == (pfx >> 20))
              atomicAdd(&lhist[(key >> 8) & 0xFFFu], 1u);
          } else {
            if ((key >> 8) == (pfx >> 8))
              atomicAdd(&lhist[key & 0xFFu], 1u);
          }
        }
      }
    }
    if (LEVEL == 0 && MOPT)
      negMask[base / 4 + (size_t)tile * (TILE / 4) + tid] = (uint8_t)nib;
  }
  if (LEVEL == 0) atomicAdd(&lpos, posc);
  __syncthreads();
  for (uint32_t i = threadIdx.x; i < BINS; i += TPB) {
    const uint32_t c = lhist[i];
    if (c) atomicAdd(&ghist[(size_t)b * BINS + i], c);
  }
  if (LEVEL == 0 && threadIdx.x == 0 && lpos != 0u)
    atomicAdd(&posCount[b], lpos);
}

// ---------------------------------------------------------------------------
// Per-sample radix-select step. Grid = B; <=4096-entry scalar scan.
// ---------------------------------------------------------------------------
template <int LEVEL>
__global__ void select_k(const int* __restrict__ label,
                         const uint32_t* __restrict__ posCount,
                         const uint32_t* __restrict__ hist,
                         uint32_t* __restrict__ prefix,
                         uint32_t* __restrict__ rankRem,
                         uint32_t* __restrict__ forceAll,
                         uint32_t* __restrict__ doneF,
                         uint32_t* __restrict__ threshKey) {
  if (threadIdx.x != 0) return;
  const int b = blockIdx.x;

  if (LEVEL == 0) {
    const uint32_t pn = posCount[b];
    const uint32_t nn = N_PER_SAMPLE - pn;
    const float ratio = c_ratios[label[b] & 7];
    if (!(ratio > 0.f)) {            // apply_ohem == False -> mask all True
      forceAll[b] = 1u;
      doneF[b] = 1u;
      threshKey[b] = 0xFFFFFFFFu;
      return;
    }
    // keep_num = min(int32(float32(pos_num)*ratio), neg_num) (truncating cast)
    long long keep = (long long)((float)pn * ratio);
    if (keep > (long long)nn) keep = (long long)nn;
    long long idx = 1 - keep;                       // raw_idx
    if (idx < 0) idx += (long long)nn;              // python negative indexing
    if (idx < 0) idx = 0;                           // clip
    if (idx > (long long)(N_PER_SAMPLE - 1)) idx = N_PER_SAMPLE - 1;
    if (idx >= (long long)nn) {                     // lands in +inf padding
      doneF[b] = 1u;
      threshKey[b] = 0xFFFFFFFFu;                   // key(x) > this never true
      return;
    }
    uint32_t rem = (uint32_t)idx;
    const uint32_t* h = hist + (size_t)b * 4096u;
    uint32_t bin = 0;
    for (; bin < 4096u; ++bin) {
      const uint32_t c = h[bin];
      if (rem < c) break;
      rem -= c;
    }
    if (bin >= 4096u) bin = 4095u;
    prefix[b] = bin << 20;
    rankRem[b] = rem;
  } else {
    if (doneF[b] | forceAll[b]) return;
    const uint32_t BINS = (LEVEL == 1) ? 4096u : 256u;
    uint32_t rem = rankRem[b];
    const uint32_t* h = hist + (size_t)b * BINS;
    uint32_t bin = 0;
    for (; bin < BINS; ++bin) {
      const uint32_t c = h[bin];
      if (rem < c) break;
      rem -= c;
    }
    if (bin >= BINS) bin = BINS - 1;
    if (LEVEL == 1) {
      prefix[b] |= bin << 8;
      rankRem[b] = rem;
    } else {
      threshKey[b] = prefix[b] | bin;               // exact threshold key
    }
  }
}

// ---------------------------------------------------------------------------
// Final masked sums: inter[b] = sum(fp*t); sums[0] += sum(fp); sums[1] += sum(t)
// (the reference's denominator is a scalar over ALL samples).
// ---------------------------------------------------------------------------
__global__ __launch_bounds__(TPB)
void final_pass(const float* __restrict__ inp, const float* __restrict__ tgt,
                const uint32_t* __restrict__ threshKey,
                const uint32_t* __restrict__ forceAll,
                float* __restrict__ inter, float* __restrict__ sums) {
  __shared__ __align__(16) float ibuf[2][TILE];
  __shared__ __align__(16) float tbuf[2][TILE];
  __shared__ float acc[3];

  const int b = blockIdx.y;
  const uint32_t tk = threshKey[b];
  const uint32_t fa = forceAll[b];
  if (threadIdx.x == 0) { acc[0] = 0.f; acc[1] = 0.f; acc[2] = 0.f; }
  __syncthreads();

  const int tid = threadIdx.x;
  const size_t base = (size_t)b * N_PER_SAMPLE +
                      (size_t)blockIdx.x * (size_t)(TILE * TILES_PER_BLOCK);

  auto issue = [&](int stage, int tile) {
    const float* gi = inp + base + (size_t)tile * TILE + tid * VEC;
    const float* gt = tgt + base + (size_t)tile * TILE + tid * VEC;
    async_load_b128((uint32_t)(uintptr_t)&ibuf[stage][tid * VEC], gi);
    async_load_b128((uint32_t)(uintptr_t)&tbuf[stage][tid * VEC], gt);
  };

  float s_i = 0.f, s_fp = 0.f, s_t = 0.f;
  issue(0, 0);
  for (int tile = 0; tile < TILES_PER_BLOCK; ++tile) {
    const int s = tile & 1;
    if (tile + 1 < TILES_PER_BLOCK) {
      issue(s ^ 1, tile + 1);
      wait_async_le2();
    } else {
      wait_async_le0();
    }
    volatile const float* iv = &ibuf[s][0];
    volatile const float* tv = &tbuf[s][0];
#pragma unroll
    for (int e = 0; e < VEC; ++e) {
      const float t = tv[tid * VEC + e];
      const float x = iv[tid * VEC + e];
      const float p = 1.f / (1.f + __expf(-x));
      const bool pos = t > 0.5f;
      const bool keep = (fa != 0u) || pos || (f2key(x) > tk);
      const float pm = keep ? p : 0.f;
      const float tm = keep ? t : 0.f;
      const float om = 1.f - pm;
      const float fp = om * om * pm;   // (1-p)^2 * p
      s_i += fp * tm;
      s_fp += fp;
      s_t += tm;
    }
  }
  atomicAdd(&acc[0], s_i);
  atomicAdd(&acc[1], s_fp);
  atomicAdd(&acc[2], s_t);
  __syncthreads();
  if (tid == 0) {
    atomicAdd(&inter[b], acc[0]);
    atomicAdd(&sums[0], acc[1]);
    atomicAdd(&sums[1], acc[2]);
  }
}

__global__ void finalize_k(const float* __restrict__ inter,
                           const float* __restrict__ sums,
                           float* __restrict__ out) {
  const int b = threadIdx.x;
  if (b < B_SAMPLES) {
    const float denom = sums[0] + sums[1] + SMOOTH_F;
    out[b] = 1.f - (2.f * inter[b] + SMOOTH_F) / denom;
  }
}

__global__ void init_ws_k(uint32_t* __restrict__ ws, int words) {
  const int i = blockIdx.x * blockDim.x + threadIdx.x;
  if (i < words) ws[i] = 0u;
}

// ---------------------------------------------------------------------------
extern "C" void kernel_launch(void* const* d_in, const int* in_sizes, int n_in,
                              void* d_out, int out_size, void* d_ws, size_t ws_size,
                              hipStream_t stream) {
  (void)in_sizes; (void)n_in; (void)out_size;
  const float* inp = (const float*)d_in[0];
  const float* tgt = (const float*)d_in[1];
  const int* label = (const int*)d_in[2];
  float* out = (float*)d_out;

  // Workspace layout (u32 words)
  uint32_t* ws = (uint32_t*)d_ws;
  uint32_t* hist1 = ws;                               // B*4096
  uint32_t* hist2 = hist1 + B_SAMPLES * 4096;         // B*4096
  uint32_t* hist3 = hist2 + B_SAMPLES * 4096;         // B*256
  uint32_t* posCount = hist3 + B_SAMPLES * 256;       // B
  uint32_t* prefix = posCount + B_SAMPLES;            // B
  uint32_t* rankRem = prefix + B_SAMPLES;             // B
  uint32_t* forceAll = rankRem + B_SAMPLES;           // B
  uint32_t* doneF = forceAll + B_SAMPLES;             // B
  uint32_t* threshKey = doneF + B_SAMPLES;            // B
  float* inter = (float*)(threshKey + B_SAMPLES);     // B
  float* sums = inter + B_SAMPLES;                    // 2: [0]=sum_fp [1]=sum_t

  const int totalWords =
      2 * B_SAMPLES * 4096 + B_SAMPLES * 256 + 6 * B_SAMPLES + B_SAMPLES + 2;

  // Optional packed neg-mask (1 byte per 4 elements): cuts refine-pass traffic.
  const size_t MASK_BYTES = (size_t)B_SAMPLES * N_PER_SAMPLE / 4;  // 4 MB
  uint8_t* negMask = (uint8_t*)(ws + totalWords);
  const bool useMask = ws_size >= (size_t)totalWords * 4 + MASK_BYTES;

  init_ws_k<<<(totalWords + TPB - 1) / TPB, TPB, 0, stream>>>(ws, totalWords);

  dim3 gData(BLOCKS_PER_SAMPLE, B_SAMPLES);

  if (useMask) {
    hist_pass<0, 1><<<gData, TPB, 0, stream>>>(inp, tgt, negMask, hist1,
                                               posCount, prefix, forceAll, doneF);
  } else {
    hist_pass<0, 0><<<gData, TPB, 0, stream>>>(inp, tgt, negMask, hist1,
                                               posCount, prefix, forceAll, doneF);
  }
  select_k<0><<<B_SAMPLES, 1, 0, stream>>>(label, posCount, hist1, prefix,
                                           rankRem, forceAll, doneF, threshKey);
  if (useMask) {
    hist_pass<1, 1><<<gData, TPB, 0, stream>>>(inp, tgt, negMask, hist2,
                                               posCount, prefix, forceAll, doneF);
  } else {
    hist_pass<1, 0><<<gData, TPB, 0, stream>>>(inp, tgt, negMask, hist2,
                                               posCount, prefix, forceAll, doneF);
  }
  select_k<1><<<B_SAMPLES, 1, 0, stream>>>(label, posCount, hist2, prefix,
                                           rankRem, forceAll, doneF, threshKey);
  if (useMask) {
    hist_pass<2, 1><<<gData, TPB, 0, stream>>>(inp, tgt, negMask, hist3,
                                               posCount, prefix, forceAll, doneF);
  } else {
    hist_pass<2, 0><<<gData, TPB, 0, stream>>>(inp, tgt, negMask, hist3,
                                               posCount, prefix, forceAll, doneF);
  }
  select_k<2><<<B_SAMPLES, 1, 0, stream>>>(label, posCount, hist3, prefix,
                                           rankRem, forceAll, doneF, threshKey);
  final_pass<<<gData, TPB, 0, stream>>>(inp, tgt, threshKey, forceAll, inter,
                                        sums);
  finalize_k<<<1, B_SAMPLES, 0, stream>>>(inter, sums, out);
}